// StructureUpdateModule_60541859004913
// MI455X (gfx1250) — compile-verified
//
#include <hip/hip_runtime.h>
#include <hip/hip_bf16.h>

// ---------------------------------------------------------------------------
// StructureUpdateModule for MI455X (gfx1250), fp32 end-to-end.
// All dense GEMMs run on V_WMMA_F32_16X16X4_F32 (full fp32 precision through
// the matrix pipe). Edge transition (dominant ~53 GFLOP) is a fused WMMA
// kernel: build e in LDS -> 512x512 -> 512x512 -> 512x128 -> LN, in place.
// Round 2: 32-edge tiles (halve L2 weight traffic) + K-pair-packed weights
// and LDS tiles so A/B fragments are single b64 loads (ds_load_b64 /
// global_load_b64) instead of two strided b32 loads.
// ---------------------------------------------------------------------------

typedef float v2f __attribute__((ext_vector_type(2)));
typedef float v8f __attribute__((ext_vector_type(8)));

#define WMMA_F32(a, b, c) \
  __builtin_amdgcn_wmma_f32_16x16x4_f32(false, (a), false, (b), (short)0, (c), false, false)

#define NB_N    1500
#define NB_K    30
#define NB_CN   384
#define NB_CZ   128
#define NB_H    12
#define NB_CHID 16
#define NB_PQ   4
#define NB_PV   8
#define CAT_W   2112   // H*(C_HID + 4*PV + C_Z) = 12*176
#define HID_E   512    // edge transition hidden = 384 + 128
#define TILE_M  32     // edges per workgroup in the fused edge kernel

// K-paired LDS tile index for a (k, row) element, row < 32, k < 512:
// pairs (k, k+1) are adjacent so an A-fragment is one aligned ds_load_b64.
__device__ __forceinline__ int pidx(int k, int row) {
  return ((k >> 1) << 6) + (row << 1) + (k & 1);
}

// ---------------------------------------------------------------------------
// Generic fp32 WMMA GEMM: C[M,N] = op(A[M,Kd] @ B[Kd,N] (+bias)), row-major.
// Block = 128 threads = 4 waves; wave w computes a 16x16 tile.
// A-frag (16x4 f32): lane l -> row (l&15), K = {k0+2*(l>>4), +1}.
// B-frag (4x16 f32): same K indices, col = col0+(l&15).
// D (16x16 f32): VGPR j -> row j + 8*(l>>4), col (l&15).
// ---------------------------------------------------------------------------
template <bool BIAS, bool RELU>
__global__ __launch_bounds__(128) void gemm_wmma_f32(
    const float* __restrict__ A, const float* __restrict__ B,
    const float* __restrict__ bias, float* __restrict__ C,
    int M, int N, int Kd) {
  const int lane = threadIdx.x & 31;
  const int wave = threadIdx.x >> 5;
  const int col0 = (blockIdx.x * 4 + wave) * 16;
  const int row0 = blockIdx.y * 16;
  if (col0 >= N) return;  // wave-uniform: EXEC stays all-ones for WMMA
  const int hl = lane >> 4;
  const int ll = lane & 15;
  int ar = row0 + ll;
  if (ar >= M) ar = M - 1;  // clamp (stores guarded; keeps EXEC full)
  const float* Arow = A + (size_t)ar * Kd;
  v8f acc = {};
  for (int k0 = 0; k0 < Kd; k0 += 4) {
    const int kk = k0 + 2 * hl;
    v2f a, b;
    a.x = Arow[kk];
    a.y = Arow[kk + 1];
    b.x = B[(size_t)kk * N + col0 + ll];
    b.y = B[(size_t)(kk + 1) * N + col0 + ll];
    acc = WMMA_F32(a, b, acc);
  }
  const float bv = BIAS ? bias[col0 + ll] : 0.f;
#pragma unroll
  for (int j = 0; j < 8; ++j) {
    const int row = row0 + j + 8 * hl;
    if (row < M) {
      float v = acc[j] + bv;
      if (RELU) v = fmaxf(v, 0.f);
      C[(size_t)row * N + col0 + ll] = v;
    }
  }
}

// ---------------------------------------------------------------------------
// Pack W[K][N] -> Wp with K-pairs adjacent: Wp[(k/2)*2N + c*2 + (k&1)].
// Makes each WMMA B-fragment a single aligned global_load_b64.
// ---------------------------------------------------------------------------
__global__ __launch_bounds__(256) void pack_kpairs(
    const float* __restrict__ W, float* __restrict__ Wp, int K, int N) {
  const int idx = blockIdx.x * 256 + threadIdx.x;
  if (idx < K * N) {
    const int k = idx / N, c = idx - k * N;
    Wp[(size_t)(k >> 1) * (2 * N) + c * 2 + (k & 1)] = W[idx];
  }
}

// ---------------------------------------------------------------------------
// Residual + LayerNorm over D=384.
// MODE 0: out = LN(x + y); MODE 1: out = LN((x + y) * mask[row]).
// In-place safe (row staged in LDS before writeback).
// ---------------------------------------------------------------------------
template <int MODE>
__global__ __launch_bounds__(128) void add_ln_kernel(
    const float* __restrict__ x, const float* __restrict__ y,
    const float* __restrict__ mask, const float* __restrict__ g,
    const float* __restrict__ b, float* __restrict__ out, int D) {
  const int row = blockIdx.x;
  __shared__ float red[128];
  __shared__ float sv[NB_CN];
  const float m = (MODE == 1) ? mask[row] : 1.f;
  float s = 0.f;
  for (int c = threadIdx.x; c < D; c += 128) {
    float v = (x[(size_t)row * D + c] + y[(size_t)row * D + c]) * m;
    sv[c] = v;
    s += v;
  }
  red[threadIdx.x] = s;
  __syncthreads();
  for (int st = 64; st > 0; st >>= 1) {
    if (threadIdx.x < st) red[threadIdx.x] += red[threadIdx.x + st];
    __syncthreads();
  }
  const float mu = red[0] / (float)D;
  __syncthreads();
  float s2 = 0.f;
  for (int c = threadIdx.x; c < D; c += 128) {
    float d = sv[c] - mu;
    s2 += d * d;
  }
  red[threadIdx.x] = s2;
  __syncthreads();
  for (int st = 64; st > 0; st >>= 1) {
    if (threadIdx.x < st) red[threadIdx.x] += red[threadIdx.x + st];
    __syncthreads();
  }
  const float rs = rsqrtf(red[0] / (float)D + 1e-5f);
  __syncthreads();
  for (int c = threadIdx.x; c < D; c += 128)
    out[(size_t)row * D + c] = (sv[c] - mu) * rs * g[c] + b[c];
}

// ---------------------------------------------------------------------------
// Edge IPA attention core: one workgroup per node (K=30 neighbors).
// ---------------------------------------------------------------------------
__global__ __launch_bounds__(256) void ipa_attn_kernel(
    const float* __restrict__ q, const float* __restrict__ kv,
    const float* __restrict__ qp, const float* __restrict__ kvp,
    const float* __restrict__ zpair, const float* __restrict__ direction,
    const float* __restrict__ rel_ori, const float* __restrict__ pair_mask,
    const int* __restrict__ eidx, const float* __restrict__ wb,
    const float* __restrict__ head_w, float* __restrict__ cat) {
  const int n = blockIdx.x;
  const int tid = threadIdx.x;
  __shared__ float sp[NB_K][144][3];  // rotated+crossed kv points (k/v mixed)
  __shared__ float sa[NB_H][NB_K];    // logits -> softmax weights
  __shared__ float sq[192], sqp[144], sdir[NB_K][3], shw[NB_H], smask[NB_K];
  __shared__ int se[NB_K];

  if (tid < 192) sq[tid] = q[(size_t)n * 192 + tid];
  if (tid < 144) sqp[tid] = qp[(size_t)n * 144 + tid];
  if (tid < NB_K * 3) sdir[tid / 3][tid % 3] = direction[(size_t)n * NB_K * 3 + tid];
  if (tid < NB_H) shw[tid] = log1pf(expf(head_w[tid])) * 0.13608276f;  // sqrt(1/54)
  if (tid < NB_K) {
    se[tid] = eidx[(size_t)n * NB_K + tid];
    smask[tid] = pair_mask[(size_t)n * NB_K + tid];
  }
  __syncthreads();

  // rot = R @ v; res = direction x rot, for all (k, h, p) points.
  for (int idx = tid; idx < NB_K * 144; idx += 256) {
    const int k = idx / 144, j = idx - k * 144;
    const int e = se[k];
    const float vx = kvp[(size_t)e * 432 + 0 * 144 + j];
    const float vy = kvp[(size_t)e * 432 + 1 * 144 + j];
    const float vz = kvp[(size_t)e * 432 + 2 * 144 + j];
    const float* R = rel_ori + ((size_t)n * NB_K + k) * 9;
    const float rx = R[0] * vx + R[1] * vy + R[2] * vz;
    const float ry = R[3] * vx + R[4] * vy + R[5] * vz;
    const float rz = R[6] * vx + R[7] * vy + R[8] * vz;
    const float dx = sdir[k][0], dy = sdir[k][1], dz = sdir[k][2];
    sp[k][j][0] = dy * rz - dz * ry;
    sp[k][j][1] = dz * rx - dx * rz;
    sp[k][j][2] = dx * ry - dy * rx;
  }
  __syncthreads();

  // logits
  for (int idx = tid; idx < NB_H * NB_K; idx += 256) {
    const int h = idx / NB_K, k = idx - h * NB_K;
    const int e = se[k];
    float qk = 0.f;
    for (int c = 0; c < NB_CHID; ++c)
      qk += sq[h * 16 + c] * kv[(size_t)e * 384 + h * 32 + c];
    qk *= 0.14433757f;  // sqrt(1/(3*16))
    float bb = 0.f;
    const float* z = zpair + ((size_t)n * NB_K + k) * NB_CZ;
    for (int c = 0; c < NB_CZ; ++c) bb += z[c] * wb[c * NB_H + h];
    bb *= 0.57735027f;  // sqrt(1/3)
    float pt = 0.f;
    for (int p = 0; p < NB_PQ; ++p) {
      const float ex = sqp[0 * 48 + h * 4 + p] - sp[k][h * 12 + p][0];
      const float ey = sqp[1 * 48 + h * 4 + p] - sp[k][h * 12 + p][1];
      const float ez = sqp[2 * 48 + h * 4 + p] - sp[k][h * 12 + p][2];
      pt += ex * ex + ey * ey + ez * ez;
    }
    pt = -0.5f * shw[h] * pt;
    sa[h][k] = qk + bb + pt + 1e8f * (smask[k] - 1.f);
  }
  __syncthreads();

  // softmax per head (K=30 is tiny; one thread per head)
  if (tid < NB_H) {
    float mx = -3.4e38f;
    for (int k = 0; k < NB_K; ++k) mx = fmaxf(mx, sa[tid][k]);
    float s = 0.f;
    for (int k = 0; k < NB_K; ++k) {
      const float eo = expf(sa[tid][k] - mx);
      sa[tid][k] = eo;
      s += eo;
    }
    const float inv = 1.f / s;
    for (int k = 0; k < NB_K; ++k) sa[tid][k] *= inv;
  }
  __syncthreads();

  float* crow = cat + (size_t)n * CAT_W;
  // o[h,c] = sum_k a * v_e   (v part of kv)
  for (int idx = tid; idx < NB_H * NB_CHID; idx += 256) {
    const int h = idx >> 4, c = idx & 15;
    float s = 0.f;
    for (int k = 0; k < NB_K; ++k)
      s += sa[h][k] * kv[(size_t)se[k] * 384 + h * 32 + 16 + c];
    crow[idx] = s;
  }
  // o_pt[h,p,:] and norms
  for (int idx = tid; idx < NB_H * NB_PV; idx += 256) {
    const int h = idx >> 3, p = idx & 7;
    float x = 0.f, y = 0.f, z = 0.f;
    for (int k = 0; k < NB_K; ++k) {
      const float a = sa[h][k];
      x += a * sp[k][h * 12 + 4 + p][0];
      y += a * sp[k][h * 12 + 4 + p][1];
      z += a * sp[k][h * 12 + 4 + p][2];
    }
    crow[192 + idx] = x;
    crow[288 + idx] = y;
    crow[384 + idx] = z;
    crow[480 + idx] = sqrtf(x * x + y * y + z * z + 1e-6f);
  }
  // o_pair[h,c] = sum_k a * z[k,c]
  for (int idx = tid; idx < NB_H * NB_CZ; idx += 256) {
    const int h = idx >> 7, c = idx & 127;
    float s = 0.f;
    const float* z = zpair + (size_t)n * NB_K * NB_CZ + c;
    for (int k = 0; k < NB_K; ++k) s += sa[h][k] * z[k * NB_CZ];
    crow[576 + idx] = s;
  }
}

// ---------------------------------------------------------------------------
// Fused edge transition: per 32-edge tile,
//   e = [pair(128) | ne_i(192) | ne_j(192)]   (K-paired layout in LDS)
//   h1 = relu(e @ wt0 + bt0); g = relu(h1 @ wt1 + bt1) + e
//   eo = g @ wf + bf; pair = LN(eo, eg, eb)   (in place)
// Block = 256 threads = 8 waves as 2 (m) x 4 (n).
// Dynamic LDS: bufA (eT->g) and bufB (h1->eo), 64KB each = 128KB/WG
// (2 WGs fit in a 320KB WGP). A-frag = one ds_load_b64; B-frag = one
// global_load_b64 from the K-pair-packed weight copies.
// ---------------------------------------------------------------------------
__global__ __launch_bounds__(256) void edge_fused_kernel(
    float* __restrict__ pair, const float* __restrict__ ne,
    const int* __restrict__ eidx, const float* __restrict__ wt0p,
    const float* __restrict__ bt0, const float* __restrict__ wt1p,
    const float* __restrict__ bt1, const float* __restrict__ wfp,
    const float* __restrict__ bf, const float* __restrict__ eg,
    const float* __restrict__ eb, int total) {
  extern __shared__ float smem[];
  float* bufA = smem;           // 16384 floats
  float* bufB = smem + 16384;   // 16384 floats
  const int tid = threadIdx.x;
  const int lane = tid & 31, wave = tid >> 5;
  const int hl = lane >> 4, ll = lane & 15;
  const int base = blockIdx.x * TILE_M;
  const int m0 = (wave >> 2) * 16;  // wave's row offset within tile

  // Build e (K-paired layout), K-dim = column of e.
  for (int idx = tid; idx < TILE_M * HID_E; idx += 256) {
    const int row = idx >> 9, col = idx & 511;
    const int edge = base + row;
    float v = 0.f;
    if (edge < total) {
      if (col < NB_CZ)
        v = pair[(size_t)edge * NB_CZ + col];
      else if (col < 320)
        v = ne[(size_t)(edge / NB_K) * 192 + (col - 128)];
      else
        v = ne[(size_t)eidx[edge] * 192 + (col - 320)];
    }
    bufA[pidx(col, row)] = v;
  }
  __syncthreads();

  // GEMM1: h1 = relu(e @ wt0 + bt0); wave covers 16 rows x 128 cols.
  {
    v8f acc[8] = {};
    const int c0 = (wave & 3) * 128;
    for (int k0 = 0; k0 < HID_E; k0 += 4) {
      const int kk = k0 + 2 * hl;
      const v2f a = *(const v2f*)(bufA + pidx(kk, m0 + ll));
#pragma unroll
      for (int t = 0; t < 8; ++t) {
        const int col = c0 + t * 16 + ll;
        const v2f b = *(const v2f*)(wt0p + (size_t)(kk >> 1) * 1024 + col * 2);
        acc[t] = WMMA_F32(a, b, acc[t]);
      }
    }
#pragma unroll
    for (int t = 0; t < 8; ++t) {
      const int col = c0 + t * 16 + ll;
      const float bv = bt0[col];
#pragma unroll
      for (int j = 0; j < 8; ++j)
        bufB[pidx(col, m0 + j + 8 * hl)] = fmaxf(acc[t][j] + bv, 0.f);
    }
  }
  __syncthreads();

  // GEMM2: g = relu(h1 @ wt1 + bt1) + e   (residual in place in bufA)
  {
    v8f acc[8] = {};
    const int c0 = (wave & 3) * 128;
    for (int k0 = 0; k0 < HID_E; k0 += 4) {
      const int kk = k0 + 2 * hl;
      const v2f a = *(const v2f*)(bufB + pidx(kk, m0 + ll));
#pragma unroll
      for (int t = 0; t < 8; ++t) {
        const int col = c0 + t * 16 + ll;
        const v2f b = *(const v2f*)(wt1p + (size_t)(kk >> 1) * 1024 + col * 2);
        acc[t] = WMMA_F32(a, b, acc[t]);
      }
    }
#pragma unroll
    for (int t = 0; t < 8; ++t) {
      const int col = c0 + t * 16 + ll;
      const float bv = bt1[col];
#pragma unroll
      for (int j = 0; j < 8; ++j) {
        const int o = pidx(col, m0 + j + 8 * hl);
        bufA[o] = fmaxf(acc[t][j] + bv, 0.f) + bufA[o];
      }
    }
  }
  __syncthreads();

  // GEMM3: eo = g @ wf + bf ; wave covers 16 rows x 32 cols of 128.
  {
    v8f acc[2] = {};
    const int c0 = (wave & 3) * 32;
    for (int k0 = 0; k0 < HID_E; k0 += 4) {
      const int kk = k0 + 2 * hl;
      const v2f a = *(const v2f*)(bufA + pidx(kk, m0 + ll));
#pragma unroll
      for (int t = 0; t < 2; ++t) {
        const int col = c0 + t * 16 + ll;
        const v2f b = *(const v2f*)(wfp + (size_t)(kk >> 1) * 256 + col * 2);
        acc[t] = WMMA_F32(a, b, acc[t]);
      }
    }
#pragma unroll
    for (int t = 0; t < 2; ++t) {
      const int col = c0 + t * 16 + ll;
      const float bv = bf[col];
#pragma unroll
      for (int j = 0; j < 8; ++j)
        bufB[(m0 + j + 8 * hl) * NB_CZ + col] = acc[t][j] + bv;
    }
  }
  __syncthreads();

  // LayerNorm over 128 per row; stats stashed in (now free) bufA.
  if (tid < TILE_M) {
    float s = 0.f, s2 = 0.f;
    for (int c = 0; c < NB_CZ; ++c) s += bufB[tid * NB_CZ + c];
    const float mu = s * (1.f / NB_CZ);
    for (int c = 0; c < NB_CZ; ++c) {
      const float d = bufB[tid * NB_CZ + c] - mu;
      s2 += d * d;
    }
    bufA[tid] = mu;
    bufA[TILE_M + tid] = rsqrtf(s2 * (1.f / NB_CZ) + 1e-5f);
  }
  __syncthreads();
  for (int idx = tid; idx < TILE_M * NB_CZ; idx += 256) {
    const int row = idx >> 7, col = idx & 127;
    const int edge = base + row;
    if (edge < total)
      pair[(size_t)edge * NB_CZ + col] =
          (bufB[idx] - bufA[row]) * bufA[TILE_M + row] * eg[col] + eb[col];
  }
}

// ---------------------------------------------------------------------------
// Host-side orchestration.
// Input flattening (setup_inputs dict order, params recursively in
// dict-literal order):
//   0 node_emb 1 pair_emb 2 rigids 3 direction 4 rel_ori 5 rigid_mask
//   6 pair_mask 7 E_idx, then per block (x2):
//   wq wkv wqp wkvp wb head_weights wo bo ln_g ln_b w1 b1 w2 b2 w3 b3 tln_g tln_b
//   then edge: wi bi wt0 bt0 wt1 bt1 wf bf eln_g eln_b
// ---------------------------------------------------------------------------
extern "C" void kernel_launch(void* const* d_in, const int* in_sizes, int n_in,
                              void* d_out, int out_size, void* d_ws,
                              size_t ws_size, hipStream_t stream) {
  const float* node_emb = (const float*)d_in[0];
  const float* pair_emb = (const float*)d_in[1];
  const float* direction = (const float*)d_in[3];
  const float* rel_ori = (const float*)d_in[4];
  const float* rigid_mask = (const float*)d_in[5];
  const float* pair_mask = (const float*)d_in[6];
  const int* E_idx = (const int*)d_in[7];

  // Workspace carve-up (floats).
  float* ws = (float*)d_ws;
  float* s     = ws;                 // 1500*384
  float* t1    = s + 576000;         // 1500*384
  float* t2    = t1 + 576000;        // 1500*384
  float* cat   = t2 + 576000;        // 1500*2112
  float* qb    = cat + 3168000;      // 1500*192
  float* kvb   = qb + 288000;        // 1500*384
  float* qpb   = kvb + 576000;       // 1500*144
  float* kvpb  = qpb + 216000;       // 1500*432
  float* neb   = kvpb + 648000;      // 1500*192
  float* ipa   = neb + 288000;       // 1500*384
  float* pc    = ipa + 576000;       // 1500*30*128 (current pair_emb)
  float* wt0p  = pc + 5760000;       // 512*512 packed
  float* wt1p  = wt0p + 262144;      // 512*512 packed
  float* wfp   = wt1p + 262144;      // 512*128 packed

  // Allow 128KB dynamic LDS for the fused edge kernel.
  (void)hipFuncSetAttribute((const void*)edge_fused_kernel,
                            hipFuncAttributeMaxDynamicSharedMemorySize,
                            2 * 16384 * (int)sizeof(float));

  hipMemcpyAsync(s, node_emb, 576000 * sizeof(float),
                 hipMemcpyDeviceToDevice, stream);
  hipMemcpyAsync(pc, pair_emb, 5760000 * sizeof(float),
                 hipMemcpyDeviceToDevice, stream);

  const dim3 blk128(128), blk256(256);
  const int Mt = (NB_N + 15) / 16;  // 94 row tiles

  for (int b = 0; b < 2; ++b) {
    void* const* P = d_in + 8 + b * 18;
    const float *wq = (const float*)P[0], *wkv = (const float*)P[1],
                *wqp = (const float*)P[2], *wkvp = (const float*)P[3],
                *wb = (const float*)P[4], *hw = (const float*)P[5],
                *wo = (const float*)P[6], *bo = (const float*)P[7],
                *lng = (const float*)P[8], *lnb = (const float*)P[9],
                *w1 = (const float*)P[10], *b1 = (const float*)P[11],
                *w2 = (const float*)P[12], *b2 = (const float*)P[13],
                *w3 = (const float*)P[14], *b3 = (const float*)P[15],
                *tlng = (const float*)P[16], *tlnb = (const float*)P[17];

    // Projections (no bias in reference IPA projections).
    gemm_wmma_f32<false, false><<<dim3(3, Mt), blk128, 0, stream>>>(
        s, wq, nullptr, qb, NB_N, 192, 384);
    gemm_wmma_f32<false, false><<<dim3(6, Mt), blk128, 0, stream>>>(
        s, wkv, nullptr, kvb, NB_N, 384, 384);
    gemm_wmma_f32<false, false><<<dim3(3, Mt), blk128, 0, stream>>>(
        s, wqp, nullptr, qpb, NB_N, 144, 384);
    gemm_wmma_f32<false, false><<<dim3(7, Mt), blk128, 0, stream>>>(
        s, wkvp, nullptr, kvpb, NB_N, 432, 384);

    ipa_attn_kernel<<<dim3(NB_N), blk256, 0, stream>>>(
        qb, kvb, qpb, kvpb, pc, direction, rel_ori, pair_mask, E_idx, wb, hw,
        cat);

    gemm_wmma_f32<true, false><<<dim3(6, Mt), blk128, 0, stream>>>(
        cat, wo, bo, ipa, NB_N, 384, CAT_W);

    // node = LN(node + ipa_out)  (in place on s)
    add_ln_kernel<0><<<dim3(NB_N), blk128, 0, stream>>>(
        s, ipa, nullptr, lng, lnb, s, NB_CN);

    // 3-layer transition
    gemm_wmma_f32<true, true><<<dim3(6, Mt), blk128, 0, stream>>>(
        s, w1, b1, t1, NB_N, 384, 384);
    gemm_wmma_f32<true, true><<<dim3(6, Mt), blk128, 0, stream>>>(
        t1, w2, b2, t2, NB_N, 384, 384);
    gemm_wmma_f32<true, false><<<dim3(6, Mt), blk128, 0, stream>>>(
        t2, w3, b3, t1, NB_N, 384, 384);

    // s = LN((t + s) * rigid_mask); final block writes straight to d_out.
    float* outp = (b == 1) ? (float*)d_out : s;
    add_ln_kernel<1><<<dim3(NB_N), blk128, 0, stream>>>(
        t1, s, rigid_mask, tlng, tlnb, outp, NB_CN);

    if (b == 0) {
      void* const* EP = d_in + 44;
      const float *wi = (const float*)EP[0], *bi = (const float*)EP[1],
                  *wt0 = (const float*)EP[2], *bt0 = (const float*)EP[3],
                  *wt1 = (const float*)EP[4], *bt1 = (const float*)EP[5],
                  *wf = (const float*)EP[6], *bf = (const float*)EP[7],
                  *elng = (const float*)EP[8], *elnb = (const float*)EP[9];

      gemm_wmma_f32<true, false><<<dim3(3, Mt), blk128, 0, stream>>>(
          s, wi, bi, neb, NB_N, 192, 384);

      // One-time K-pair packing of edge weights (B-frags -> single b64).
      pack_kpairs<<<dim3(1024), blk256, 0, stream>>>(wt0, wt0p, 512, 512);
      pack_kpairs<<<dim3(1024), blk256, 0, stream>>>(wt1, wt1p, 512, 512);
      pack_kpairs<<<dim3(256), blk256, 0, stream>>>(wf, wfp, 512, 128);

      const int total = NB_N * NB_K;  // 45000 edges
      edge_fused_kernel<<<dim3((total + TILE_M - 1) / TILE_M), blk256,
                          2 * 16384 * sizeof(float), stream>>>(
          pc, neb, E_idx, wt0p, bt0, wt1p, bt1, wfp, bf, elng, elnb, total);
    }
  }
  (void)in_sizes; (void)n_in; (void)out_size; (void)ws_size;
}